// ScaledDotProductAttention_19688130085313
// MI455X (gfx1250) — compile-verified
//
#include <hip/hip_runtime.h>
#include <stdint.h>

// Problem constants (reference: B=2, H=16, S=2048, D=64, fp32)
#define BATCH 2
#define HEADS 16
#define SEQ   2048
#define DHEAD 64
#define SCALE 0.125f        // 1/sqrt(64)
#define LROW  2064          // padded logits row stride (2064 % 64 == 16)
#define KROW  68            // padded K-tile row stride (68*4B = 272B, 16B aligned; 68%64==4
                            //  -> conflict-free b64 fragment reads across both lane halves)

typedef float v2f __attribute__((ext_vector_type(2)));
typedef float v8f __attribute__((ext_vector_type(8)));
typedef int   v4i __attribute__((vector_size(16)));   // matches async-LDS builtin pointee

#define AS1 __attribute__((address_space(1)))
#define AS3 __attribute__((address_space(3)))

#if __has_builtin(__builtin_amdgcn_global_load_async_to_lds_b128)
#define USE_ASYNC 1
#else
#define USE_ASYNC 0
#endif

template <int N>
__device__ __forceinline__ void wait_async() {
#if USE_ASYNC
#if __has_builtin(__builtin_amdgcn_s_wait_asynccnt)
    __builtin_amdgcn_s_wait_asynccnt(N);
#else
    if constexpr (N == 0) asm volatile("s_wait_asynccnt 0x0" ::: "memory");
    else                  asm volatile("s_wait_asynccnt 0x8" ::: "memory");
#endif
#endif
}

// Stage one 16x64 f32 K tile (4KB, contiguous in global) into a padded LDS buffer.
// Global side: fully coalesced b128; LDS side: rows padded to KROW floats.
__device__ __forceinline__ void stage_k_tile(const float4* __restrict__ gsrc4,
                                             float* __restrict__ buf, int lane) {
#pragma unroll
    for (int j = 0; j < 8; ++j) {
        const int idx = lane + 32 * j;      // float4 index 0..255 over the tile
        const int row = idx >> 4;           // key within tile
        const int c4  = idx & 15;           // float4 column
        float* dst = buf + row * KROW + c4 * 4;
#if USE_ASYNC
        __builtin_amdgcn_global_load_async_to_lds_b128(
            (AS1 v4i*)(gsrc4 + idx), (AS3 v4i*)dst, 0, 0);
#else
        *(float4*)dst = gsrc4[idx];
#endif
    }
}

// One workgroup = one (b, h, 16-query-row tile). 256 threads = 8 wave32.
__global__ __launch_bounds__(256, 1)
void attn_fused_kernel(const float* __restrict__ q,
                       const float* __restrict__ k,
                       const float* __restrict__ v,
                       const uint8_t* __restrict__ mask,
                       float* __restrict__ out,      // [B,H,S,D]
                       float* __restrict__ wout) {   // [B,H,S,S]
    __shared__ float lds_logits[16][LROW];      // 129.0 KB: 16 q-rows x 2048 logits
    __shared__ float lds_k[8][2][16 * KROW];    //  68.0 KB: per-wave double-buffered K tiles
    __shared__ float lds_q[16][DHEAD];          //   4.0 KB: Q tile
    __shared__ float lds_red[16][16];           //   1.0 KB: softmax reductions

    const int qt = blockIdx.x;         // 0..127
    const int h  = blockIdx.y;         // 0..15
    const int b  = blockIdx.z;         // 0..1
    const int q0 = qt * 16;

    const int tid  = threadIdx.x;      // 0..255
    const int lane = tid & 31;
    const int wave = tid >> 5;         // 0..7
    const int half = lane >> 4;        // K-pair select within WMMA frag
    const int lm   = lane & 15;        // M (A) / N (B,C) within tile

    const size_t bh    = (size_t)b * HEADS + h;
    const float* Qbase = q + (bh * SEQ + q0) * DHEAD;   // 16x64, contiguous
    const float* Kbase = k + bh * SEQ * DHEAD;
    const float* Vbase = v + bh * SEQ * DHEAD;

    // ---- stage Q tile to LDS (1024 floats, coalesced) ----
    for (int i = tid; i < 16 * DHEAD; i += 256)
        lds_q[i >> 6][i & 63] = Qbase[i];
    __syncthreads();

    // ---- pass 1: logits = scale * Q @ K^T ----
    // Each wave owns 256 consecutive keys = 16 tiles; K tiles are DMA-staged into
    // LDS (double buffered, ASYNCcnt-pipelined) so global reads are pure b128.
    {
        float* kbuf0 = &lds_k[wave][0][0];
        float* kbuf1 = &lds_k[wave][1][0];
        const float4* ktiles = (const float4*)(Kbase + (size_t)(wave * 256) * DHEAD);

        stage_k_tile(ktiles, kbuf0, lane);              // prefetch tile 0

        const v2f* qrow = (const v2f*)&lds_q[lm][0];
        for (int kt = 0; kt < 16; ++kt) {
            float* curbuf = (kt & 1) ? kbuf1 : kbuf0;
            if (kt < 15) {
#if USE_ASYNC
                // prior ds reads of the target buffer must be complete
                asm volatile("s_wait_dscnt 0x0" ::: "memory");
#endif
                stage_k_tile(ktiles + (kt + 1) * 256,   // next 4KB tile
                             (kt & 1) ? kbuf0 : kbuf1, lane);
                wait_async<8>();                        // current tile's 8 DMAs done
            } else {
                wait_async<0>();
            }

            const v2f* krow = (const v2f*)(curbuf + lm * KROW);
            v8f acc = {};
#pragma unroll
            for (int s = 0; s < 16; ++s) {              // K-steps of 4 over D=64
                v2f a  = qrow[2 * s + half];            // A: Q[lm][4s + 2*half .. +1]
                v2f bb = krow[2 * s + half];            // B: K[key][4s + 2*half .. +1] (LDS)
                acc = __builtin_amdgcn_wmma_f32_16x16x4_f32(
                          false, a, false, bb, (short)0, acc, false, false);
            }
            const int keybase = wave * 256 + kt * 16;
#pragma unroll
            for (int i = 0; i < 8; ++i)                 // C: row = i + 8*half, col = key
                lds_logits[i + 8 * half][keybase + lm] = acc[i] * SCALE;
        }
    }
    __syncthreads();

    // ---- softmax over each of 16 rows (16 threads/row, strided/coalesced) ----
    {
        const int r   = tid >> 4;
        const int sub = tid & 15;
        const uint8_t* mrow = mask + ((size_t)b * SEQ + (q0 + r)) * SEQ;
        float* lrow = &lds_logits[r][0];

        // masked fill (faithful -1e-9) + local max
        float mloc = -3.4e38f;
#pragma unroll 4
        for (int j = 0; j < 128; ++j) {
            const int c = sub + (j << 4);
            float val = lrow[c];
            if (mrow[c]) val = -1e-9f;
            lrow[c] = val;
            mloc = fmaxf(mloc, val);
        }
        lds_red[r][sub] = mloc;
        __syncthreads();
        float rmax = lds_red[r][0];
#pragma unroll
        for (int i = 1; i < 16; ++i) rmax = fmaxf(rmax, lds_red[r][i]);
        __syncthreads();

        // exp + local sum
        float sloc = 0.0f;
#pragma unroll 4
        for (int j = 0; j < 128; ++j) {
            const int c = sub + (j << 4);
            float e = __expf(lrow[c] - rmax);
            lrow[c] = e;
            sloc += e;
        }
        lds_red[r][sub] = sloc;
        __syncthreads();
        float rsum = 0.0f;
#pragma unroll
        for (int i = 0; i < 16; ++i) rsum += lds_red[r][i];
        const float inv = 1.0f / rsum;

        // normalize in LDS + stream weights to HBM (the dominant traffic)
        float* wrow = wout + (bh * SEQ + (q0 + r)) * SEQ;
#pragma unroll 4
        for (int j = 0; j < 128; ++j) {
            const int c = sub + (j << 4);
            float w = lrow[c] * inv;
            lrow[c] = w;
            wrow[c] = w;
        }
    }
    __syncthreads();

    // ---- pass 2: O = P @ V  (each wave: its 256-key slice, 4 n-tiles over D=64) ----
    v8f acc2[4];
#pragma unroll
    for (int nt = 0; nt < 4; ++nt) acc2[nt] = (v8f){};
    {
        const int kb0 = wave * 256;
        for (int s = 0; s < 64; ++s) {              // K-steps of 4 over the 256-key slice
            const int k0 = kb0 + 4 * s;
            v2f a;                                  // A frag: P[lm][k0 + 2*half .. +1]
            a.x = lds_logits[lm][k0 + 2 * half];
            a.y = lds_logits[lm][k0 + 2 * half + 1];
            const float* vr0 = Vbase + (size_t)(k0 + 2 * half) * DHEAD;
#pragma unroll
            for (int nt = 0; nt < 4; ++nt) {
                const int col = nt * 16 + lm;
                v2f bb;
                bb.x = vr0[col];                    // V[k0+2*half  ][col] (64B coalesced)
                bb.y = vr0[DHEAD + col];            // V[k0+2*half+1][col]
                acc2[nt] = __builtin_amdgcn_wmma_f32_16x16x4_f32(
                               false, a, false, bb, (short)0, acc2[nt], false, false);
            }
        }
    }
    __syncthreads();   // all waves done reading P before LDS reuse

    // ---- cross-wave reduction of 16x64 partials through LDS ----
    {
        float* part = &lds_logits[0][0];            // reuse: 8 waves * 1024 floats = 32KB
#pragma unroll
        for (int i = 0; i < 8; ++i) {
            const int row = i + 8 * half;
#pragma unroll
            for (int nt = 0; nt < 4; ++nt)
                part[wave * 1024 + row * DHEAD + nt * 16 + lm] = acc2[nt][i];
        }
        __syncthreads();

        float* obase = out + (bh * SEQ + q0) * DHEAD;   // 16x64, contiguous
        for (int e = tid; e < 16 * DHEAD; e += 256) {
            float sum = 0.0f;
#pragma unroll
            for (int w = 0; w < 8; ++w) sum += part[w * 1024 + e];
            obase[e] = sum;
        }
    }
}

extern "C" void kernel_launch(void* const* d_in, const int* in_sizes, int n_in,
                              void* d_out, int out_size, void* d_ws, size_t ws_size,
                              hipStream_t stream) {
    const float*   q    = (const float*)d_in[0];
    const float*   k    = (const float*)d_in[1];
    const float*   v    = (const float*)d_in[2];
    const uint8_t* mask = (const uint8_t*)d_in[3];   // jnp.bool_ -> 1 byte/elem

    float* out  = (float*)d_out;                                   // [B,H,S,D] first
    float* wout = out + (size_t)BATCH * HEADS * SEQ * DHEAD;       // then [B,H,S,S]

    dim3 grid(SEQ / 16, HEADS, BATCH);   // (128, 16, 2)
    dim3 block(256);
    attn_fused_kernel<<<grid, block, 0, stream>>>(q, k, v, mask, out, wout);
}